// CropPoolLayer_39505109188895
// MI455X (gfx1250) — compile-verified
//
#include <hip/hip_runtime.h>

// Problem constants (from reference setup)
#define HFM   64          // feature map height
#define WFM   64          // feature map width
#define CFM   512         // channels
#define CROPN 14          // pre-pool crop size
#define POOLN 7           // pooled output size
#define PATCH 18          // max staged patch rows/cols (box <=256px / stride16 -> <=17+1)
#define CT    32          // channels per block tile
#define G4    (CT/4)      // float4 groups per pixel within tile = 8
#define NTHREADS 128      // 4 waves (wave32)

#define USE_TDM 1         // 1: stage via tensor_load_to_lds (TDM); 0: per-lane async copies

typedef __attribute__((ext_vector_type(4))) unsigned int v4u;
typedef __attribute__((ext_vector_type(8))) unsigned int v8u;

__device__ __forceinline__ float4 lerp4(float4 a, float4 b, float w) {
    return make_float4(a.x + (b.x - a.x) * w,
                       a.y + (b.y - a.y) * w,
                       a.z + (b.z - a.z) * w,
                       a.w + (b.w - a.w) * w);
}
__device__ __forceinline__ float4 max4(float4 a, float4 b) {
    return make_float4(fmaxf(a.x, b.x), fmaxf(a.y, b.y),
                       fmaxf(a.z, b.z), fmaxf(a.w, b.w));
}
__device__ __forceinline__ unsigned rfl(unsigned x) {
    return (unsigned)__builtin_amdgcn_readfirstlane((int)x);
}

__global__ __launch_bounds__(NTHREADS)
void crop_resize_pool_kernel(const float* __restrict__ bottom,
                             const float* __restrict__ rois,
                             const float* __restrict__ im_info,
                             float* __restrict__ out)
{
    // Staged ROI patch: 18*18 pixels * 8 float4 (32 channels) = 41472 B LDS
    __shared__ float4 patch[PATCH * PATCH * G4];
    __shared__ int   sy0[CROPN], sy1[CROPN], sx0[CROPN], sx1[CROPN];
    __shared__ float sly[CROPN], slx[CROPN];
    __shared__ int   svy[CROPN], svx[CROPN];
    __shared__ int   sbid, srow0, scol0, sph, spw, suselds;

    const int r     = blockIdx.x;          // ROI index
    const int cbase = blockIdx.y * CT;     // channel tile base
    const int t     = threadIdx.x;

    // --- per-ROI sampling grid setup (28 lanes) ----------------------------
    if (t < 2 * CROPN) {
        const bool isy = (t < CROPN);
        const int  i   = isy ? t : (t - CROPN);
        const float im_h = im_info[0];
        const float im_w = im_info[1];
        const float inv  = isy ? (1.0f / im_h) : (1.0f / im_w);
        const float a    = rois[(size_t)r * 5 + (isy ? 2 : 1)] * inv; // y1 / x1
        const float b    = rois[(size_t)r * 5 + (isy ? 4 : 3)] * inv; // y2 / x2
        const float dim1 = (float)((isy ? HFM : WFM) - 1);
        // TF crop_and_resize grid: s = a*(D-1) + i*(b-a)*(D-1)/(crop-1)
        const float s = a * dim1 + (float)i * ((b - a) * dim1 / (float)(CROPN - 1));
        const float f = floorf(s);
        int lo = (int)f;
        int hi = (int)ceilf(s);
        lo = min(max(lo, 0), (int)dim1);
        hi = min(max(hi, 0), (int)dim1);
        const int v = (s >= 0.0f && s <= dim1) ? 1 : 0;
        if (isy) { sy0[i] = lo; sy1[i] = hi; sly[i] = s - f; svy[i] = v; }
        else     { sx0[i] = lo; sx1[i] = hi; slx[i] = s - f; svx[i] = v; }
        if (t == 0) sbid = (int)rois[(size_t)r * 5 + 0];
    }
    __syncthreads();
    if (t == 0) {
        // grid coords are affine in i -> extremes at endpoints
        const int r0 = min(sy0[0], sy0[CROPN - 1]);
        const int r1 = max(sy1[0], sy1[CROPN - 1]);
        const int c0 = min(sx0[0], sx0[CROPN - 1]);
        const int c1 = max(sx1[0], sx1[CROPN - 1]);
        srow0 = r0; scol0 = c0;
        sph = r1 - r0 + 1; spw = c1 - c0 + 1;
        suselds = (sph <= PATCH && spw <= PATCH) ? 1 : 0;
    }
    __syncthreads();

    const int  bid    = sbid;
    const int  row0   = srow0, col0 = scol0;
    const int  ph     = sph,   pw   = spw;
    const bool uselds = (suselds != 0);

#if USE_TDM
    // --- stage patch into LDS with one Tensor Data Mover descriptor --------
    // 3D tile: dim0 = 32 channels (contiguous), dim1 = pw cols (stride C),
    // dim2 = ph rows (stride W*C). LDS dest is compacted pixel-major, which
    // matches patch[] layout exactly (128 B per pixel).
    if (uselds) {
        const int waveid = __builtin_amdgcn_readfirstlane((int)t) >> 5;
        if (waveid == 0) {  // wave 0 issues the DMA (TDM ignores EXEC)
            const unsigned long long ga = (unsigned long long)(size_t)
                (bottom + (((size_t)(bid * HFM + row0)) * WFM + (size_t)col0) * CFM
                        + (size_t)cbase);
            // low 32 bits of flat shared pointer == LDS byte offset (ISA 10.2)
            const unsigned ldsb = rfl((unsigned)(size_t)(&patch[0]));
            const unsigned galo = rfl((unsigned)(ga & 0xFFFFFFFFu));
            const unsigned gahi = rfl((unsigned)(ga >> 32));
            const unsigned upw  = rfl((unsigned)pw);
            const unsigned uph  = rfl((unsigned)ph);

            v4u g0;  // D# group 0
            g0.x = 1u;                                  // count=1 (valid), user mode
            g0.y = ldsb;                                // lds_addr[31:0]
            g0.z = galo;                                // global_addr[31:0]
            g0.w = (gahi & 0x01FFFFFFu) | 0x80000000u;  // global_addr[56:32], type=2

            v8u g1;  // D# group 1
            g1.s0 = (2u << 16);                         // data_size=4B, wg_mask=0
            g1.s1 = ((unsigned)CT << 16);               // tensor_dim0[15:0]=32
            g1.s2 = (upw << 16);                        // tensor_dim0 hi=0, tensor_dim1 lo=pw
            g1.s3 = ((unsigned)CT << 16);               // tensor_dim1 hi=0, tile_dim0=32
            g1.s4 = upw | (uph << 16);                  // tile_dim1=pw, tile_dim2=ph
            g1.s5 = (unsigned)CFM;                      // tensor_dim0_stride=512
            g1.s6 = ((unsigned)(WFM * CFM) & 0xFFFFu) << 16; // dim1_stride lo16 (0x8000)
            g1.s7 = (unsigned)((WFM * CFM) >> 16);      // dim1_stride[47:16] = 0

            v4u g2;  // D# group 2
            g2.x = uph;                                 // tensor_dim2=ph
            g2.y = 0u; g2.z = 0u; g2.w = 0u;            // tensor_dim3 / tile_dim3 unused
            v4u g3;  // D# group 3 (unused dims)
            g3.x = 0u; g3.y = 0u; g3.z = 0u; g3.w = 0u;

            asm volatile("tensor_load_to_lds %0, %1, %2, %3"
                         :: "s"(g0), "s"(g1), "s"(g2), "s"(g3)
                         : "memory");
#if defined(__has_builtin) && __has_builtin(__builtin_amdgcn_s_wait_tensorcnt)
            __builtin_amdgcn_s_wait_tensorcnt(0);
#else
            asm volatile("s_wait_tensorcnt 0x0" ::: "memory");
#endif
        }
    }
    __syncthreads();
#else
    // --- stage patch with per-lane async global->LDS copies ----------------
    if (uselds) {
        const int npix = ph * pw;
        const int gsl  = t % G4;
        for (int p = t / G4; p < npix; p += NTHREADS / G4) {
            const int py = p / pw;
            const int px = p - py * pw;
            const size_t goff =
                (((size_t)(bid * HFM + row0 + py)) * WFM + (size_t)(col0 + px)) * CFM
                + (size_t)(cbase + 4 * gsl);
            const unsigned long long gaddr =
                (unsigned long long)(size_t)(bottom + goff);
            const unsigned ldsoff = (unsigned)(size_t)(&patch[p * G4 + gsl]);
            asm volatile("global_load_async_to_lds_b128 %0, %1, off"
                         :: "v"(ldsoff), "v"(gaddr)
                         : "memory");
        }
        asm volatile("s_wait_asynccnt 0x0" ::: "memory");
    }
    __syncthreads();
#endif

    // --- sample + 2x2 max pool ---------------------------------------------
    const int gs = t % G4;
    const int c  = cbase + 4 * gs;
    for (int cell = t / G4; cell < POOLN * POOLN; cell += NTHREADS / G4) {
        const int pyc = cell / POOLN;
        const int pxc = cell - pyc * POOLN;
        float4 acc = make_float4(-3.402823466e38f, -3.402823466e38f,
                                 -3.402823466e38f, -3.402823466e38f);
        #pragma unroll
        for (int dy = 0; dy < 2; ++dy) {
            #pragma unroll
            for (int dx = 0; dx < 2; ++dx) {
                const int i = 2 * pyc + dy;
                const int j = 2 * pxc + dx;
                const float ly = sly[i];
                const float lx = slx[j];
                const int y0 = sy0[i], y1 = sy1[i];
                const int x0 = sx0[j], x1 = sx1[j];
                float4 tl, tr, bl, br;
                if (uselds) {
                    const int ra = (y0 - row0) * pw;
                    const int rb = (y1 - row0) * pw;
                    const int ca = x0 - col0;
                    const int cb = x1 - col0;
                    tl = patch[(ra + ca) * G4 + gs];
                    tr = patch[(ra + cb) * G4 + gs];
                    bl = patch[(rb + ca) * G4 + gs];
                    br = patch[(rb + cb) * G4 + gs];
                } else {
                    const float* base = bottom + (size_t)bid * HFM * WFM * CFM + c;
                    tl = *(const float4*)(base + ((size_t)y0 * WFM + x0) * CFM);
                    tr = *(const float4*)(base + ((size_t)y0 * WFM + x1) * CFM);
                    bl = *(const float4*)(base + ((size_t)y1 * WFM + x0) * CFM);
                    br = *(const float4*)(base + ((size_t)y1 * WFM + x1) * CFM);
                }
                const float4 top = lerp4(tl, tr, lx);
                const float4 bot = lerp4(bl, br, lx);
                float4 v = lerp4(top, bot, ly);
                if (!(svy[i] && svx[j])) v = make_float4(0.f, 0.f, 0.f, 0.f);
                acc = max4(acc, v);
            }
        }
        float4* dst = (float4*)(out +
            (((size_t)r * POOLN + pyc) * POOLN + pxc) * CFM + c);
        *dst = acc;
    }
}

extern "C" void kernel_launch(void* const* d_in, const int* in_sizes, int n_in,
                              void* d_out, int out_size, void* d_ws, size_t ws_size,
                              hipStream_t stream) {
    (void)n_in; (void)out_size; (void)d_ws; (void)ws_size;
    const float* bottom  = (const float*)d_in[0];
    const float* rois    = (const float*)d_in[1];
    const float* im_info = (const float*)d_in[2];
    float* out = (float*)d_out;
    const int nroi = in_sizes[1] / 5;
    dim3 grid(nroi, CFM / CT);   // (512 ROIs, 16 channel tiles of 32)
    crop_resize_pool_kernel<<<grid, NTHREADS, 0, stream>>>(bottom, rois, im_info, out);
}